// FourierFFTLayer_13657996001424
// MI455X (gfx1250) — compile-verified
//
#include <hip/hip_runtime.h>

// Fused FFT->IFFT (identity-up-to-roundoff) on (8,4096,1024) f32 rows.
// Single pass over HBM (~268 MB => ~11.5 us floor at 23.3 TB/s).
// 1024 = 16 * (16 * 4) Cooley-Tukey; the two DFT-16 stages run on the f32
// matrix pipe via V_WMMA_F32_16X16X4_F32 (K=16 as 4 chained K=4 WMMAs),
// twiddles + radix-4 on VALU. One wave (wave32) owns one row in LDS.

typedef float v2f __attribute__((ext_vector_type(2)));
typedef float v8f __attribute__((ext_vector_type(8)));

#define WAVES_PER_BLOCK 2
#define THREADS (WAVES_PER_BLOCK * 32)
#define FFT_N 1024
#define TWO_PI 6.28318530717958647692f

__device__ __forceinline__ v8f wmma_f32(v2f a, v2f b, v8f c) {
  // (neg_a, A, neg_b, B, c_mod, C, reuse_a, reuse_b) -- f32 16x16x4
  return __builtin_amdgcn_wmma_f32_16x16x4_f32(false, a, false, b, (short)0, c,
                                               false, false);
}

// DFT-16 along K as a 16x16 matrix applied to 64 columns, on the matrix pipe.
// MODE 0: data is [n1(16) x n2(64)], element (row,col) at row*64+col.
// MODE 1: per-k1 sub-matrices [m1(16) x m2(4)] at base k1*64 + m1*4 + m2;
//         WMMA tile columns enumerate (k1,m2) pairs (4 k1-slabs per tile).
// Complex: Y = (C + i*s*S) * X  with s = -1 fwd, +1 inv.
// In-place safe per tile: a tile's loads and stores touch the same column set
// and all loads precede all stores (DS ops are in-order per wave).
template <int MODE>
__device__ __forceinline__ void dft16_stage(float* Re, float* Im,
                                            const float* Fc, const float* Fs,
                                            float s) {
  const int lane = threadIdx.x & 31;
  const int half = lane >> 4;   // 0: lanes 0-15, 1: lanes 16-31
  const int l16  = lane & 15;
#pragma unroll
  for (int tile = 0; tile < 4; ++tile) {
    const int ct      = tile * 16 + l16;                          // B/C/D column
    const int caddr   = (MODE == 0) ? ct : ((ct >> 2) * 64 + (ct & 3));
    const int rstride = (MODE == 0) ? 64 : 4;
    v8f accR = {0.f, 0.f, 0.f, 0.f, 0.f, 0.f, 0.f, 0.f};
    v8f accI = {0.f, 0.f, 0.f, 0.f, 0.f, 0.f, 0.f, 0.f};
#pragma unroll
    for (int c = 0; c < 4; ++c) {
      // A fragment (16x4, 2 VGPRs): VGPR j holds K = 4c + 2*half + j, M = l16.
      const int k0 = 4 * c + 2 * half;
      const int fo = l16 * 16 + k0;
      v2f aC  = {Fc[fo], Fc[fo + 1]};
      v2f aSp = {s * Fs[fo], s * Fs[fo + 1]};   // +s*S
      v2f aSn = {-aSp.x, -aSp.y};               // -s*S
      // B fragment (4x16, 2 VGPRs): VGPR j holds row K = 4c + 2*half + j.
      const int b0 = caddr + k0 * rstride;
      v2f bR = {Re[b0], Re[b0 + rstride]};
      v2f bI = {Im[b0], Im[b0 + rstride]};
      // Yre = C*Xr - s*S*Xi ; Yim = C*Xi + s*S*Xr
      accR = wmma_f32(aC,  bR, accR);
      accR = wmma_f32(aSn, bI, accR);
      accI = wmma_f32(aC,  bI, accI);
      accI = wmma_f32(aSp, bR, accI);
    }
    // D (16x16, 8 VGPRs): VGPR r holds rows r (lanes 0-15) and r+8 (16-31).
#pragma unroll
    for (int r = 0; r < 8; ++r) {
      const int m = r + 8 * half;
      Re[caddr + m * rstride] = accR[r];
      Im[caddr + m * rstride] = accI[r];
    }
  }
}

// Four-step inter-stage twiddle: A[k1][n2] *= exp(i*s*2pi*k1*n2/1024).
__device__ __forceinline__ void twiddle1024(float* Re, float* Im, float s) {
  const int lane = threadIdx.x & 31;
#pragma unroll 4
  for (int idx = lane; idx < FFT_N; idx += 32) {
    const int k1 = idx >> 6, n2 = idx & 63;
    const float ang = s * (TWO_PI / 1024.0f) * (float)(k1 * n2);
    float sn, cs;
    __sincosf(ang, &sn, &cs);
    const float re = Re[idx], im = Im[idx];
    Re[idx] = re * cs - im * sn;
    Im[idx] = re * sn + im * cs;
  }
}

// W64 twiddle + radix-4 + digit-reversal scatter: src[k1][j1][m2] ->
// dst[k1 + 16*j1 + 256*j2] (natural-order output).
__device__ __forceinline__ void radix4_final(const float* sRe, const float* sIm,
                                             float* dRe, float* dIm,
                                             float s, float scale) {
  const int lane = threadIdx.x & 31;
#pragma unroll
  for (int g = lane; g < 256; g += 32) {
    const int k1 = g >> 4, j1 = g & 15;
    const int base = k1 * 64 + j1 * 4;
    float tr[4], ti[4];
#pragma unroll
    for (int m2 = 0; m2 < 4; ++m2) {
      const float re = sRe[base + m2], im = sIm[base + m2];
      const float ang = s * (TWO_PI / 64.0f) * (float)(j1 * m2);
      float sn, cs;
      __sincosf(ang, &sn, &cs);
      tr[m2] = re * cs - im * sn;
      ti[m2] = re * sn + im * cs;
    }
    const float u0r = tr[0] + tr[2], u0i = ti[0] + ti[2];
    const float u1r = tr[0] - tr[2], u1i = ti[0] - ti[2];
    const float u2r = tr[1] + tr[3], u2i = ti[1] + ti[3];
    const float u3r = tr[1] - tr[3], u3i = ti[1] - ti[3];
    const int o = k1 + 16 * j1;
    dRe[o      ] = (u0r + u2r) * scale;  dIm[o      ] = (u0i + u2i) * scale;
    dRe[o + 256] = (u1r - s * u3i) * scale;  dIm[o + 256] = (u1i + s * u3r) * scale;
    dRe[o + 512] = (u0r - u2r) * scale;  dIm[o + 512] = (u0i - u2i) * scale;
    dRe[o + 768] = (u1r + s * u3i) * scale;  dIm[o + 768] = (u1i - s * u3r) * scale;
  }
}

__global__ void __launch_bounds__(THREADS)
fft_ifft_rows_kernel(const float* __restrict__ x, float* __restrict__ out) {
  __shared__ float Fc[256], Fs[256];                       // DFT-16 matrix
  __shared__ float pRe[WAVES_PER_BLOCK][2][FFT_N];         // ping-pong planes
  __shared__ float pIm[WAVES_PER_BLOCK][2][FFT_N];

  const int tid = threadIdx.x;
  for (int i = tid; i < 256; i += THREADS) {
    const float ang = (TWO_PI / 16.0f) * (float)((i >> 4) * (i & 15));
    float sn, cs;
    __sincosf(ang, &sn, &cs);
    Fc[i] = cs;
    Fs[i] = sn;
  }

  const int wave = tid >> 5;
  const int lane = tid & 31;
  const long long row = (long long)blockIdx.x * WAVES_PER_BLOCK + wave;
  const float* __restrict__ xr = x + row * FFT_N;
  float* p0R = pRe[wave][0]; float* p0I = pIm[wave][0];
  float* p1R = pRe[wave][1]; float* p1I = pIm[wave][1];

  // Load one real row (b128 per lane), zero imaginary plane.
  for (int v = lane; v < FFT_N / 4; v += 32) {
    const float4 t = ((const float4*)xr)[v];
    p0R[v * 4 + 0] = t.x; p0R[v * 4 + 1] = t.y;
    p0R[v * 4 + 2] = t.z; p0R[v * 4 + 3] = t.w;
    p0I[v * 4 + 0] = 0.f; p0I[v * 4 + 1] = 0.f;
    p0I[v * 4 + 2] = 0.f; p0I[v * 4 + 3] = 0.f;
  }
  __syncthreads();

  // Forward FFT (s = -1): plane0 -> plane1 (natural order both ends).
  dft16_stage<0>(p0R, p0I, Fc, Fs, -1.0f);  __syncthreads();
  twiddle1024(p0R, p0I, -1.0f);             __syncthreads();
  dft16_stage<1>(p0R, p0I, Fc, Fs, -1.0f);  __syncthreads();
  radix4_final(p0R, p0I, p1R, p1I, -1.0f, 1.0f);  __syncthreads();

  // Inverse FFT (s = +1): plane1 -> plane0, scale 1/N at the end.
  dft16_stage<0>(p1R, p1I, Fc, Fs, 1.0f);   __syncthreads();
  twiddle1024(p1R, p1I, 1.0f);              __syncthreads();
  dft16_stage<1>(p1R, p1I, Fc, Fs, 1.0f);   __syncthreads();
  radix4_final(p1R, p1I, p0R, p0I, 1.0f, 1.0f / 1024.0f);  __syncthreads();

  // Store real part (b128 per lane).
  float* __restrict__ orow = out + row * FFT_N;
  for (int v = lane; v < FFT_N / 4; v += 32) {
    float4 t;
    t.x = p0R[v * 4 + 0]; t.y = p0R[v * 4 + 1];
    t.z = p0R[v * 4 + 2]; t.w = p0R[v * 4 + 3];
    ((float4*)orow)[v] = t;
  }
}

extern "C" void kernel_launch(void* const* d_in, const int* in_sizes, int n_in,
                              void* d_out, int out_size, void* d_ws, size_t ws_size,
                              hipStream_t stream) {
  (void)n_in; (void)out_size; (void)d_ws; (void)ws_size;
  const float* x = (const float*)d_in[0];
  float* out = (float*)d_out;
  const int nrows = in_sizes[0] / FFT_N;                 // 8*4096 = 32768
  const int nblocks = (nrows + WAVES_PER_BLOCK - 1) / WAVES_PER_BLOCK;
  fft_ifft_rows_kernel<<<dim3(nblocks), dim3(THREADS), 0, stream>>>(x, out);
}